// MambaClassifier_45071386804325
// MI455X (gfx1250) — compile-verified
//
#include <hip/hip_runtime.h>
#include <cmath>

#define D_MODEL   256
#define N_LAYERS  4
#define D_INNER   512
#define D_STATE   16
#define D_CONV    4
#define DT_RANK   16
#define NUM_CLS   10
#define BATCH     8
#define SEQLEN    1024
#define ROWS      (BATCH * SEQLEN)   /* 8192 */

#define KCHUNK    128                /* K-columns staged in LDS per round  */
#define KPAD      4                  /* row pad (dwords): bank spread + 16B align */

typedef float v2f __attribute__((ext_vector_type(2)));
typedef float v8f __attribute__((ext_vector_type(8)));
typedef int   v4i __attribute__((vector_size(4 * sizeof(int))));

enum { EPI_NONE = 0, EPI_BIAS = 1, EPI_BIAS_SOFTPLUS = 2, EPI_RESID_BIAS = 3 };

__device__ __forceinline__ float silu_f(float x) { return x / (1.0f + expf(-x)); }

// --------------------------------------------------------------------------
// Async global->LDS staging (CDNA5 GLOBAL_LOAD_ASYNC_TO_LDS_B128, ASYNCcnt),
// with a synchronous fallback if the builtin is not declared.
// Probe-derived signature: (v4i AS1*, v4i AS3*, imm offset, imm cpol).
// --------------------------------------------------------------------------
#if __has_builtin(__builtin_amdgcn_global_load_async_to_lds_b128)
#define HAVE_ASYNC_LDS 1
#else
#define HAVE_ASYNC_LDS 0
#endif

__device__ __forceinline__ void async_copy_b128(const float* gsrc, float* ldst)
{
#if HAVE_ASYNC_LDS
    float* g = const_cast<float*>(gsrc);
    __builtin_amdgcn_global_load_async_to_lds_b128(
        (__attribute__((address_space(1))) v4i*)g,
        (__attribute__((address_space(3))) v4i*)ldst, 0, 0);
#else
    *(float4*)ldst = *(const float4*)gsrc;
#endif
}

__device__ __forceinline__ void wait_async_all()
{
#if HAVE_ASYNC_LDS
#if __has_builtin(__builtin_amdgcn_s_wait_asynccnt)
    __builtin_amdgcn_s_wait_asynccnt(0);
#else
    asm volatile("s_wait_asynccnt 0x0" ::: "memory");
#endif
#endif
}

// ---------------------------------------------------------------------------
// fp32 WMMA GEMM with LDS-staged weights:
//   C(M,N) = A(M,K) * W(N,K)^T  [+ bias / softplus / resid]
// Block = 8 waves = 128 rows x (16*NT) cols. The W strip is staged into LDS in
// 128-wide K-chunks (async-to-LDS), shared by all 8 waves; each wave then runs
// K-steps of V_WMMA_F32_16X16X4_F32 with A fragments from global (b64) and W
// fragments from LDS (ds_load_b64, conflict-free via KPAD row padding).
// Grid is exact, staging divergence reconverges before the K-loop, so EXEC is
// all-ones at every WMMA.
// ---------------------------------------------------------------------------
template <int NT>
__global__ __launch_bounds__(256) void gemm_wmma_lds(
    const float* __restrict__ A, int lda,
    const float* __restrict__ W, int ldw,
    const float* __restrict__ bias,
    const float* __restrict__ resid, int ldr,
    float* __restrict__ C, int ldc,
    int M, int N, int K, int epi)
{
    extern __shared__ float lw[];                 // (16*NT) x (KCHUNK+KPAD)
    const int wave = threadIdx.x >> 5;
    const int lane = threadIdx.x & 31;
    const int groupsN = N / (16 * NT);
    const int mblk = blockIdx.x / groupsN;        // 128-row block index
    const int n0   = (blockIdx.x % groupsN) * (16 * NT);
    const int m0   = mblk * 128 + wave * 16;
    const int half = lane >> 4;                   // K half select (0: K=0..1, 1: K=2..3)
    const int lidx = lane & 15;                   // M for A-frag, N for B-frag
    const int ldsStride = KCHUNK + KPAD;          // 132 dwords (528B, 16B aligned)

    v8f acc[NT];
#pragma unroll
    for (int t = 0; t < NT; ++t) acc[t] = (v8f){};

    for (int kc = 0; kc < K; kc += KCHUNK) {
        const int ck = (K - kc < KCHUNK) ? (K - kc) : KCHUNK;

        // ---- cooperative async stage of W[n0 .. n0+16*NT) x [kc .. kc+ck) ----
        const int vecsPerRow = ck >> 2;                  // b128 transfers per row
        const int vecs = (16 * NT) * vecsPerRow;
        for (int idx = threadIdx.x; idx < vecs; idx += 256) {
            const int r = idx / vecsPerRow;
            const int c = (idx % vecsPerRow) << 2;
            async_copy_b128(W + (size_t)(n0 + r) * ldw + kc + c,
                            &lw[r * ldsStride + c]);
        }
        wait_async_all();
        __syncthreads();

        // ---- 16x16x4 WMMA K-steps over this chunk ----
        const float* arow = A + (size_t)(m0 + lidx) * lda + kc + 2 * half;
        const float* wrow = &lw[lidx * ldsStride + 2 * half];
        for (int k = 0; k < ck; k += 4) {
            v2f a;
            a.x = arow[k]; a.y = arow[k + 1];            // A[m][kc+k+2h .. +1]
#pragma unroll
            for (int t = 0; t < NT; ++t) {
                const float* wr = wrow + (16 * t) * ldsStride;
                v2f b;
                b.x = wr[k]; b.y = wr[k + 1];            // B[k+2h][n] = W[n][kc+k+2h]
                acc[t] = __builtin_amdgcn_wmma_f32_16x16x4_f32(
                    /*neg_a=*/false, a, /*neg_b=*/false, b,
                    /*c_mod=*/(short)0, acc[t], /*reuse_a=*/false, /*reuse_b=*/false);
            }
        }
        __syncthreads();                                 // LDS reuse fence
    }

#pragma unroll
    for (int t = 0; t < NT; ++t) {
        const int col = n0 + 16 * t + lidx;
        const float bv = (epi == EPI_NONE) ? 0.0f : bias[col];
#pragma unroll
        for (int j = 0; j < 8; ++j) {
            const int row = m0 + j + 8 * half;           // C/D: VGPR j -> M=j (+8 upper half)
            float v = acc[t][j] + bv;
            if (epi == EPI_BIAS_SOFTPLUS)   v = (v > 20.0f) ? v : log1pf(expf(v));
            else if (epi == EPI_RESID_BIAS) v += resid[(size_t)row * ldr + col];
            C[(size_t)row * ldc + col] = v;
        }
    }
}

// ---------------------------------------------------------------------------
// LayerNorm over D_MODEL=256; one row per block.
// ---------------------------------------------------------------------------
__global__ __launch_bounds__(256) void layernorm_k(
    const float* __restrict__ in, const float* __restrict__ w,
    const float* __restrict__ b, float* __restrict__ out)
{
    __shared__ float red[256];
    const int row = blockIdx.x;
    const int d = threadIdx.x;
    const float v = in[(size_t)row * D_MODEL + d];
    red[d] = v; __syncthreads();
    for (int s = 128; s > 0; s >>= 1) { if (d < s) red[d] += red[d + s]; __syncthreads(); }
    const float mu = red[0] * (1.0f / D_MODEL);
    __syncthreads();
    const float c = v - mu;
    red[d] = c * c; __syncthreads();
    for (int s = 128; s > 0; s >>= 1) { if (d < s) red[d] += red[d + s]; __syncthreads(); }
    const float var = red[0] * (1.0f / D_MODEL);
    out[(size_t)row * D_MODEL + d] = c * rsqrtf(var + 1e-5f) * w[d] + b[d];
}

// ---------------------------------------------------------------------------
// Causal depthwise conv (K=4) over the u-half of xz, fused SiLU.
// ---------------------------------------------------------------------------
__global__ void conv_silu_k(const float* __restrict__ xz,
                            const float* __restrict__ cw,
                            const float* __restrict__ cb,
                            float* __restrict__ uo)
{
    const int idx = blockIdx.x * blockDim.x + threadIdx.x;
    if (idx >= ROWS * D_INNER) return;
    const int d = idx % D_INNER;
    const int l = (idx / D_INNER) % SEQLEN;
    const int b = idx / (D_INNER * SEQLEN);
    float acc = cb[d];
#pragma unroll
    for (int j = 0; j < D_CONV; ++j) {
        const int p = l - (D_CONV - 1) + j;
        if (p >= 0)
            acc += cw[d * D_CONV + j] * xz[((size_t)(b * SEQLEN + p)) * (2 * D_INNER) + d];
    }
    uo[idx] = silu_f(acc);
}

// ---------------------------------------------------------------------------
// Selective scan, fused with output gating y *= silu(res).
// Thread = one (d_local, n) state; 16 d x 16 n per block; serial over L.
// Only the n==0 lane of each 16-lane group loads delta/u/res (then shfl
// broadcast); y is reduced over n with shfl_xor inside the 16-lane group.
// ---------------------------------------------------------------------------
__global__ __launch_bounds__(256) void scan_gate_k(
    const float* __restrict__ delta,   // (B,L,Din)
    const float* __restrict__ u,       // (B,L,Din)
    const float* __restrict__ xdbl,    // (B,L,48): [dt | Bm | Cm]
    const float* __restrict__ xz,      // (B,L,2*Din): res = xz[..., Din:]
    const float* __restrict__ A_log,   // (Din,16) for this layer
    const float* __restrict__ Dp,      // (Din)
    float* __restrict__ y)             // (B,L,Din)
{
    const int b    = blockIdx.x >> 5;            // 32 blocks per batch element
    const int dblk = blockIdx.x & 31;
    const int lane = threadIdx.x & 31;
    const int n    = lane & 15;
    const int dl   = threadIdx.x >> 4;
    const int d    = dblk * 16 + dl;
    const int gb   = lane & 16;                  // first lane of this 16-lane group
    const float Adn = -expf(A_log[d * D_STATE + n]);
    const float Dd  = Dp[d];
    float h = 0.0f;
    const size_t baseBL = (size_t)b * SEQLEN;
    for (int t = 0; t < SEQLEN; ++t) {
        const size_t bl = baseBL + t;
        float dt = 0.0f, ut = 0.0f, res = 0.0f;
        if (n == 0) {
            dt  = delta[bl * D_INNER + d];
            ut  = u[bl * D_INNER + d];
            res = xz[bl * (2 * D_INNER) + D_INNER + d];
        }
        dt = __shfl(dt, gb, 32);
        ut = __shfl(ut, gb, 32);
        const float Bt = xdbl[bl * (DT_RANK + 2 * D_STATE) + DT_RANK + n];
        const float Ct = xdbl[bl * (DT_RANK + 2 * D_STATE) + DT_RANK + D_STATE + n];
        h = expf(dt * Adn) * h + dt * Bt * ut;
        float yv = h * Ct;
        yv += __shfl_xor(yv, 1, 32);
        yv += __shfl_xor(yv, 2, 32);
        yv += __shfl_xor(yv, 4, 32);
        yv += __shfl_xor(yv, 8, 32);
        if (n == 0) y[bl * D_INNER + d] = (yv + ut * Dd) * silu_f(res);
    }
}

// ---------------------------------------------------------------------------
// h[bl,d] = x[bl] * input_w[d] + input_b[d]
// ---------------------------------------------------------------------------
__global__ void input_proj_k(const float* __restrict__ x,
                             const float* __restrict__ iw,
                             const float* __restrict__ ib,
                             float* __restrict__ h)
{
    const int idx = blockIdx.x * blockDim.x + threadIdx.x;
    if (idx >= ROWS * D_MODEL) return;
    const int d = idx % D_MODEL;
    const int bl = idx / D_MODEL;
    h[idx] = x[bl] * iw[d] + ib[d];
}

// ---------------------------------------------------------------------------
// Mean-pool over L then tiny classifier (8 x 10).
// ---------------------------------------------------------------------------
__global__ __launch_bounds__(256) void pool_cls_k(const float* __restrict__ hn,
                                                  const float* __restrict__ cw,
                                                  const float* __restrict__ cb,
                                                  float* __restrict__ out)
{
    __shared__ float pooled[D_MODEL];
    const int b = blockIdx.x;
    const int d = threadIdx.x;
    float s = 0.0f;
    for (int l = 0; l < SEQLEN; ++l)
        s += hn[((size_t)(b * SEQLEN + l)) * D_MODEL + d];
    pooled[d] = s * (1.0f / SEQLEN);
    __syncthreads();
    if (d < NUM_CLS) {
        float acc = cb[d];
        for (int k = 0; k < D_MODEL; ++k) acc += pooled[k] * cw[d * D_MODEL + k];
        out[b * NUM_CLS + d] = acc;
    }
}

// ---------------------------------------------------------------------------
extern "C" void kernel_launch(void* const* d_in, const int* in_sizes, int n_in,
                              void* d_out, int out_size, void* d_ws, size_t ws_size,
                              hipStream_t stream)
{
    const float* x         = (const float*)d_in[0];
    const float* input_w   = (const float*)d_in[1];
    const float* input_b   = (const float*)d_in[2];
    const float* ln_w      = (const float*)d_in[3];
    const float* ln_b      = (const float*)d_in[4];
    const float* in_proj_w = (const float*)d_in[5];
    const float* in_proj_b = (const float*)d_in[6];
    const float* conv_w    = (const float*)d_in[7];
    const float* conv_b    = (const float*)d_in[8];
    const float* x_proj_w  = (const float*)d_in[9];
    const float* dt_proj_w = (const float*)d_in[10];
    const float* dt_proj_b = (const float*)d_in[11];
    const float* A_log     = (const float*)d_in[12];
    const float* Dparam    = (const float*)d_in[13];
    const float* out_proj_w= (const float*)d_in[14];
    const float* out_proj_b= (const float*)d_in[15];
    const float* norm_w    = (const float*)d_in[16];
    const float* norm_b    = (const float*)d_in[17];
    const float* cls_w     = (const float*)d_in[18];
    const float* cls_b     = (const float*)d_in[19];
    (void)in_sizes; (void)n_in; (void)out_size; (void)ws_size;

    // Workspace layout (floats), ~102 MB total.
    float* ws = (float*)d_ws;
    float* h  = ws;                                   // (ROWS, 256)
    float* hn = h  + (size_t)ROWS * D_MODEL;          // (ROWS, 256)
    float* xz = hn + (size_t)ROWS * D_MODEL;          // (ROWS, 1024)
    float* uc = xz + (size_t)ROWS * 2 * D_INNER;      // (ROWS, 512)
    float* xd = uc + (size_t)ROWS * D_INNER;          // (ROWS, 48)
    float* dl = xd + (size_t)ROWS * (DT_RANK + 2 * D_STATE); // (ROWS, 512)
    float* yb = dl + (size_t)ROWS * D_INNER;          // (ROWS, 512)

    const size_t sh4 = (size_t)(16 * 4) * (KCHUNK + KPAD) * sizeof(float); // 33,792 B
    const size_t sh3 = (size_t)(16 * 3) * (KCHUNK + KPAD) * sizeof(float); // 25,344 B

    input_proj_k<<<(ROWS * D_MODEL + 255) / 256, 256, 0, stream>>>(x, input_w, input_b, h);

    for (int i = 0; i < N_LAYERS; ++i) {
        layernorm_k<<<ROWS, 256, 0, stream>>>(h, ln_w + i * D_MODEL, ln_b + i * D_MODEL, hn);

        // in_proj: (8192,256) x (1024,256)^T -> xz (8192,1024), +bias. 128x64 blocks.
        {
            const int blocks = (ROWS / 128) * ((2 * D_INNER) / 64);  // 1024
            gemm_wmma_lds<4><<<blocks, 256, sh4, stream>>>(
                hn, D_MODEL,
                in_proj_w + (size_t)i * 2 * D_INNER * D_MODEL, D_MODEL,
                in_proj_b + i * 2 * D_INNER, nullptr, 0,
                xz, 2 * D_INNER, ROWS, 2 * D_INNER, D_MODEL, EPI_BIAS);
        }

        conv_silu_k<<<(ROWS * D_INNER + 255) / 256, 256, 0, stream>>>(
            xz, conv_w + (size_t)i * D_INNER * D_CONV, conv_b + i * D_INNER, uc);

        // x_proj: (8192,512) x (48,512)^T -> xd (8192,48), no bias. 128x48 blocks.
        {
            const int blocks = (ROWS / 128) * (48 / 48);             // 64
            gemm_wmma_lds<3><<<blocks, 256, sh3, stream>>>(
                uc, D_INNER,
                x_proj_w + (size_t)i * 48 * D_INNER, D_INNER,
                nullptr, nullptr, 0,
                xd, 48, ROWS, 48, D_INNER, EPI_NONE);
        }

        // dt_proj: dt = xd[:, :16] (lda=48), (8192,16) x (512,16)^T -> delta, softplus(+bias).
        {
            const int blocks = (ROWS / 128) * (D_INNER / 64);        // 512
            gemm_wmma_lds<4><<<blocks, 256, sh4, stream>>>(
                xd, DT_RANK + 2 * D_STATE,
                dt_proj_w + (size_t)i * D_INNER * DT_RANK, DT_RANK,
                dt_proj_b + i * D_INNER, nullptr, 0,
                dl, D_INNER, ROWS, D_INNER, DT_RANK, EPI_BIAS_SOFTPLUS);
        }

        scan_gate_k<<<BATCH * (D_INNER / 16), 256, 0, stream>>>(
            dl, uc, xd, xz,
            A_log + (size_t)i * D_INNER * D_STATE, Dparam + i * D_INNER, yb);

        // out_proj: (8192,512) x (256,512)^T + h + bias -> h (in place, 1:1 element map).
        {
            const int blocks = (ROWS / 128) * (D_MODEL / 64);        // 256
            gemm_wmma_lds<4><<<blocks, 256, sh4, stream>>>(
                yb, D_INNER,
                out_proj_w + (size_t)i * D_MODEL * D_INNER, D_INNER,
                out_proj_b + i * D_MODEL, h, D_MODEL,
                h, D_MODEL, ROWS, D_MODEL, D_INNER, EPI_RESID_BIAS);
        }
    }

    layernorm_k<<<ROWS, 256, 0, stream>>>(h, norm_w, norm_b, hn);
    pool_cls_k<<<BATCH, 256, 0, stream>>>(hn, cls_w, cls_b, (float*)d_out);
}